// CalibratedNormFixedAlpha_76192719831877
// MI455X (gfx1250) — compile-verified
//
#include <hip/hip_runtime.h>

#define HW        12544      // 112*112
#define C_DIM     64
#define B_DIM     64
#define NPLANES   (B_DIM * C_DIM)   // 4096 (b,c) planes
#define NGROUPS   32
#define EPSV      1e-5f

typedef float v2f __attribute__((ext_vector_type(2)));
typedef float v4f __attribute__((ext_vector_type(4)));
typedef float v8f __attribute__((ext_vector_type(8)));

// ---------------------------------------------------------------------------
// Pass 1: per-(b,c) plane sum and sum-of-squares via V_WMMA_F32_16X16X4_F32.
// One wave (32 threads) per plane; each iteration loads 128 consecutive
// floats (b128 per lane) and feeds them as two 16x4 A-tiles into WMMA with an
// all-ones B matrix: D = A*1 + C accumulates row sums (exact f32 RNE).
// ---------------------------------------------------------------------------
__global__ __launch_bounds__(32) void cnfa_stats_kernel(
    const float* __restrict__ x,
    float* __restrict__ sum_out,
    float* __restrict__ sq_out)
{
    const int plane = blockIdx.x;
    const int lane  = threadIdx.x;
    const float* p  = x + (size_t)plane * HW;

    v8f accS = {};            // accumulates sum(x)
    v8f accQ = {};            // accumulates sum(x*x)
    const v2f ones = {1.0f, 1.0f};   // B matrix = all ones (4x16)

    #pragma unroll 2
    for (int i = 0; i < HW / 128; ++i) {          // 98 iterations
        v4f v = *(const v4f*)(p + i * 128 + lane * 4);
        v2f a0 = {v.x, v.y};
        v2f a1 = {v.z, v.w};
        v2f q0 = a0 * a0;                          // packed f32 mults
        v2f q1 = a1 * a1;
        accS = __builtin_amdgcn_wmma_f32_16x16x4_f32(false, a0, false, ones,
                                                     (short)0, accS, false, false);
        accS = __builtin_amdgcn_wmma_f32_16x16x4_f32(false, a1, false, ones,
                                                     (short)0, accS, false, false);
        accQ = __builtin_amdgcn_wmma_f32_16x16x4_f32(false, q0, false, ones,
                                                     (short)0, accQ, false, false);
        accQ = __builtin_amdgcn_wmma_f32_16x16x4_f32(false, q1, false, ones,
                                                     (short)0, accQ, false, false);
    }

    // All 16 columns of D are identical (B = ones). Lanes 0-15 hold rows
    // M=0..7 in VGPR 0..7, lanes 16-31 hold rows M=8..15. Per-lane partial:
    float s = accS[0] + accS[1] + accS[2] + accS[3] +
              accS[4] + accS[5] + accS[6] + accS[7];
    float q = accQ[0] + accQ[1] + accQ[2] + accQ[3] +
              accQ[4] + accQ[5] + accQ[6] + accQ[7];
    // Fold the two half-wave row groups:
    s += __shfl_xor(s, 16, 32);
    q += __shfl_xor(q, 16, 32);

    if (lane == 0) {
        sum_out[plane] = s;
        sq_out[plane]  = q;
    }
}

// ---------------------------------------------------------------------------
// Pass 2: fold global-BN stats (sums over B of per-plane sums) and
// per-sample-BN stats + label-routed group affine into per-plane (A, B):
//   out = x * A[b,c] + B[b,c]
// with A = (1-a)*g_g*inv_g + a*g_s*inv_s,  B likewise. ALPHA = 0.5.
// ---------------------------------------------------------------------------
__global__ __launch_bounds__(256) void cnfa_coeff_kernel(
    const float* __restrict__ S,       // [4096] per-plane sum
    const float* __restrict__ Q,       // [4096] per-plane sumsq
    const float* __restrict__ gamma_g, // [64]
    const float* __restrict__ beta_g,  // [64]
    const float* __restrict__ gamma_grp, // [32*64]
    const float* __restrict__ beta_grp,  // [32*64]
    const int*   __restrict__ labels,    // [64]
    float* __restrict__ Aout,
    float* __restrict__ Bout)
{
    __shared__ float ag[C_DIM];
    __shared__ float bg[C_DIM];

    const int tid = threadIdx.x;

    if (tid < C_DIM) {
        float ssum = 0.0f, qsum = 0.0f;
        #pragma unroll 8
        for (int b = 0; b < B_DIM; ++b) {
            ssum += S[b * C_DIM + tid];
            qsum += Q[b * C_DIM + tid];
        }
        const float invN = 1.0f / (float)(B_DIM * HW);
        const float mu   = ssum * invN;
        const float var  = qsum * invN - mu * mu;   // biased
        const float a    = gamma_g[tid] * __frsqrt_rn(var + EPSV);
        ag[tid] = a;
        bg[tid] = beta_g[tid] - mu * a;
    }
    __syncthreads();

    const float invHW = 1.0f / (float)HW;
    for (int idx = tid; idx < NPLANES; idx += 256) {
        const int b = idx >> 6;
        const int c = idx & (C_DIM - 1);
        const float mu  = S[idx] * invHW;
        const float var = Q[idx] * invHW - mu * mu; // biased
        const int   gi  = labels[b] % NGROUPS;      // labels >= 0
        const float a_s = gamma_grp[gi * C_DIM + c] * __frsqrt_rn(var + EPSV);
        const float b_s = beta_grp[gi * C_DIM + c] - mu * a_s;
        Aout[idx] = 0.5f * (ag[c] + a_s);           // ALPHA = 0.5
        Bout[idx] = 0.5f * (bg[c] + b_s);
    }
}

// ---------------------------------------------------------------------------
// Pass 3: out = fma(x, A[plane], B[plane]).  x re-read is last-use (the
// first pass left it L2-resident; 192 MB L2 vs 205 MB tensor) -> NT load;
// output is write-once -> NT store so it doesn't evict x from L2.
// ---------------------------------------------------------------------------
__global__ __launch_bounds__(256) void cnfa_apply_kernel(
    const float* __restrict__ x,
    const float* __restrict__ A,
    const float* __restrict__ B,
    float* __restrict__ out)
{
    const int plane = blockIdx.y;
    const int t = blockIdx.x * 256 + threadIdx.x;   // float4 index in plane
    if (t >= HW / 4) return;

    const float a = A[plane];
    const float b = B[plane];

    const v4f* xp = (const v4f*)(x + (size_t)plane * HW) + t;
    v4f*       op = (v4f*)(out + (size_t)plane * HW) + t;

    v4f v = __builtin_nontemporal_load(xp);
    v4f o = v * a + b;
    __builtin_nontemporal_store(o, op);
}

// ---------------------------------------------------------------------------
extern "C" void kernel_launch(void* const* d_in, const int* in_sizes, int n_in,
                              void* d_out, int out_size, void* d_ws, size_t ws_size,
                              hipStream_t stream)
{
    const float* x         = (const float*)d_in[0];
    const float* gamma_g   = (const float*)d_in[1];
    const float* beta_g    = (const float*)d_in[2];
    const float* gamma_grp = (const float*)d_in[3];
    const float* beta_grp  = (const float*)d_in[4];
    const int*   labels    = (const int*)d_in[5];
    float*       out       = (float*)d_out;

    float* ws = (float*)d_ws;          // needs 4*4096 floats = 64 KB
    float* S  = ws;
    float* Q  = ws + NPLANES;
    float* A  = ws + 2 * NPLANES;
    float* B  = ws + 3 * NPLANES;

    cnfa_stats_kernel<<<NPLANES, 32, 0, stream>>>(x, S, Q);
    cnfa_coeff_kernel<<<1, 256, 0, stream>>>(S, Q, gamma_g, beta_g,
                                             gamma_grp, beta_grp, labels, A, B);
    dim3 grid3((HW / 4 + 255) / 256, NPLANES, 1);   // 13 x 4096
    cnfa_apply_kernel<<<grid3, 256, 0, stream>>>(x, A, B, out);
}